// HetAgg_2576980377820
// MI455X (gfx1250) — compile-verified
//
#include <hip/hip_runtime.h>
#include <math.h>

// Problem constants (match reference)
#define Bq 8192
#define Sq 10
#define Tq 3
#define Nq 65536
#define Dq 128
#define Lq 2
#define NEG_SLOPE 0.01f
#define PAD 132   // LDS row pitch in floats -> conflict-free 16x4 A-fragment b64 loads

typedef float v2f __attribute__((ext_vector_type(2)));
typedef float v4f __attribute__((ext_vector_type(4)));
typedef float v8f __attribute__((ext_vector_type(8)));

// ---------------------------------------------------------------------------
// RNN aggregation: for each (layer l, type t, 16-row batch tile):
//   h_s = tanh(x_s @ Wih^T + bih + h_{s-1} @ Whh^T + bhh),  agg = mean_s h_s
// 8 waves per block, wave w owns output columns [16w, 16w+16).
// Weight B-fragments are register-resident; x_s and h live in padded LDS.
// ---------------------------------------------------------------------------
__launch_bounds__(256, 1)
__global__ void rnn_agg_kernel(const float* __restrict__ emb,    // [T,N,D]
                               const float* __restrict__ Wih,    // [L,T,D,D]
                               const float* __restrict__ Whh,    // [L,T,D,D]
                               const float* __restrict__ bih,    // [L,T,D]
                               const float* __restrict__ bhh,    // [L,T,D]
                               const int*   __restrict__ neigh,  // [T,B,S]
                               float*       __restrict__ agg)    // [L,T,B,D]
{
    __shared__ float xbuf[2][16 * PAD];
    __shared__ float hbuf[16 * PAD];

    const int mblock = blockIdx.x;
    const int t      = blockIdx.y;
    const int l      = blockIdx.z;
    const int tid    = threadIdx.x;
    const int lane   = tid & 31;
    const int wave   = tid >> 5;
    const int n0     = wave * 16;     // output-column base for this wave
    const int lane15 = lane & 15;
    const int half   = lane >> 4;
    const int gb     = mblock * 16;   // global batch row base
    const int lt     = l * Tq + t;

    // ---- B-operand weight fragments into registers -------------------------
    // WMMA 16x16x4 f32 B layout: vgpr0 lanes0-15 = K=k0 row, lanes16-31 = K=k0+2;
    // vgpr1 = K=k0+1 / k0+3.  B[d][e] = W[e][d], so lane holds W[row n][k0+2*half .. +1].
    const float* wih_row = Wih + ((size_t)(lt * Dq + n0 + lane15)) * Dq;
    const float* whh_row = Whh + ((size_t)(lt * Dq + n0 + lane15)) * Dq;
    v2f fih[32], fhh[32];
#pragma unroll
    for (int j = 0; j < 32; ++j) {
        v4f q = *(const v4f*)(wih_row + 4 * j);
        v4f r = *(const v4f*)(whh_row + 4 * j);
        v2f a, b;
        if (half) { a.x = q.z; a.y = q.w; b.x = r.z; b.y = r.w; }
        else      { a.x = q.x; a.y = q.y; b.x = r.x; b.y = r.y; }
        fih[j] = a;
        fhh[j] = b;
    }
    const float bn = bih[lt * Dq + n0 + lane15] + bhh[lt * Dq + n0 + lane15];

    // ---- zero hidden state -------------------------------------------------
    for (int i = tid; i < 16 * PAD; i += 256) hbuf[i] = 0.0f;

    // ---- gather mapping: 16 threads per row, 8 floats per thread -----------
    const int gm   = tid >> 4;
    const int gcol = (tid & 15) * 8;
    {
        int idx = neigh[((size_t)t * Bq + gb + gm) * Sq + 0];
        const float* src = emb + ((size_t)t * Nq + idx) * Dq + gcol;
        *(v4f*)&xbuf[0][gm * PAD + gcol]     = *(const v4f*)(src);
        *(v4f*)&xbuf[0][gm * PAD + gcol + 4] = *(const v4f*)(src + 4);
    }
    __syncthreads();

    // A-fragment / C-layout lane mapping
    const int am   = lane15;          // A: M = lane&15 for both halves
    const int koff = half * 2;        // A: K sub-offset
    const int crow = half * 8;        // C: rows v (+0) or v+8 (+8)
    const int ncol = n0 + lane15;     // C: column

    float hs[8];
#pragma unroll
    for (int v = 0; v < 8; ++v) hs[v] = 0.0f;

    for (int s = 0; s < Sq; ++s) {
        const int cb = s & 1;

        // prefetch next step's gathered rows into registers (hidden under WMMA)
        v4f pa = {}, pb = {};
        const bool more = (s + 1 < Sq);
        if (more) {
            int idx = neigh[((size_t)t * Bq + gb + gm) * Sq + (s + 1)];
            const float* src = emb + ((size_t)t * Nq + idx) * Dq + gcol;
            pa = *(const v4f*)(src);
            pb = *(const v4f*)(src + 4);
        }

        // pre = x_s @ Wih^T + h @ Whh^T : two interleaved WMMA accumulate chains
        v8f cx = {};
        v8f ch = {};
#pragma unroll
        for (int j = 0; j < 32; ++j) {
            v2f ax = *(const v2f*)&xbuf[cb][am * PAD + 4 * j + koff];
            v2f ah = *(const v2f*)&hbuf[am * PAD + 4 * j + koff];
            cx = __builtin_amdgcn_wmma_f32_16x16x4_f32(false, ax, false, fih[j],
                                                       (short)0, cx, false, false);
            ch = __builtin_amdgcn_wmma_f32_16x16x4_f32(false, ah, false, fhh[j],
                                                       (short)0, ch, false, false);
        }

        float hnew[8];
#pragma unroll
        for (int v = 0; v < 8; ++v) {
            hnew[v] = tanhf(cx[v] + ch[v] + bn);
            hs[v] += hnew[v];
        }

        __syncthreads();   // all waves finished reading hbuf / xbuf[cb]
#pragma unroll
        for (int v = 0; v < 8; ++v)
            hbuf[(v + crow) * PAD + ncol] = hnew[v];
        if (more) {
            *(v4f*)&xbuf[1 - cb][gm * PAD + gcol]     = pa;
            *(v4f*)&xbuf[1 - cb][gm * PAD + gcol + 4] = pb;
        }
        __syncthreads();   // new h / x visible
    }

    // agg = mean over S hidden states
    const float inv = 1.0f / (float)Sq;
#pragma unroll
    for (int v = 0; v < 8; ++v) {
        agg[((size_t)lt * Bq + gb + v + crow) * Dq + ncol] = hs[v] * inv;
    }
}

// ---------------------------------------------------------------------------
// Attention + leaky ReLU for one layer. One wave per batch row (4 floats/lane).
// Layer 0 fuses the cur0 = emb[0][ids] gather (cur_in == nullptr).
// ---------------------------------------------------------------------------
__launch_bounds__(256, 1)
__global__ void att_kernel(const float* __restrict__ cur_in,  // [B,D] or null
                           const float* __restrict__ emb,     // [T,N,D] (type 0 used)
                           const int*   __restrict__ ids,     // [B] or null
                           const float* __restrict__ agg_l,   // [T,B,D]
                           const float* __restrict__ watt,    // [2D]
                           float*       __restrict__ out)     // [B,D]
{
    const int lane = threadIdx.x & 31;
    const int wave = threadIdx.x >> 5;
    const int b    = blockIdx.x * 8 + wave;
    const int d0   = lane * 4;

    v4f w1 = *(const v4f*)(watt + d0);
    v4f w2 = *(const v4f*)(watt + Dq + d0);

    v4f cand[4];
    if (cur_in) cand[0] = *(const v4f*)(cur_in + (size_t)b * Dq + d0);
    else        cand[0] = *(const v4f*)(emb + ((size_t)ids[b]) * Dq + d0);
#pragma unroll
    for (int t = 0; t < Tq; ++t)
        cand[1 + t] = *(const v4f*)(agg_l + ((size_t)t * Bq + b) * Dq + d0);

    // partial dot products: p0 = cur.w1 ; p[1+i] = cand_i.w2
    float p[5];
    p[0] = cand[0].x * w1.x + cand[0].y * w1.y + cand[0].z * w1.z + cand[0].w * w1.w;
#pragma unroll
    for (int i = 0; i < 4; ++i)
        p[1 + i] = cand[i].x * w2.x + cand[i].y * w2.y + cand[i].z * w2.z + cand[i].w * w2.w;

#pragma unroll
    for (int off = 16; off >= 1; off >>= 1) {
#pragma unroll
        for (int i = 0; i < 5; ++i) p[i] += __shfl_xor(p[i], off, 32);
    }

    float sc[4];
#pragma unroll
    for (int i = 0; i < 4; ++i) sc[i] = p[0] + p[1 + i];
    float mx = fmaxf(fmaxf(sc[0], sc[1]), fmaxf(sc[2], sc[3]));
    float e[4], se = 0.0f;
#pragma unroll
    for (int i = 0; i < 4; ++i) { e[i] = __expf(sc[i] - mx); se += e[i]; }
    const float is = 1.0f / se;

    v4f o = {};
#pragma unroll
    for (int i = 0; i < 4; ++i) {
        float wgt = e[i] * is;
        o.x += wgt * cand[i].x;
        o.y += wgt * cand[i].y;
        o.z += wgt * cand[i].z;
        o.w += wgt * cand[i].w;
    }
    o.x = o.x > 0.0f ? o.x : NEG_SLOPE * o.x;
    o.y = o.y > 0.0f ? o.y : NEG_SLOPE * o.y;
    o.z = o.z > 0.0f ? o.z : NEG_SLOPE * o.z;
    o.w = o.w > 0.0f ? o.w : NEG_SLOPE * o.w;

    *(v4f*)(out + (size_t)b * Dq + d0) = o;
}

// ---------------------------------------------------------------------------
extern "C" void kernel_launch(void* const* d_in, const int* in_sizes, int n_in,
                              void* d_out, int out_size, void* d_ws, size_t ws_size,
                              hipStream_t stream) {
    const float* emb  = (const float*)d_in[0];   // [T,N,D]
    const float* Wih  = (const float*)d_in[1];   // [L,T,D,D]
    const float* Whh  = (const float*)d_in[2];   // [L,T,D,D]
    const float* bih  = (const float*)d_in[3];   // [L,T,D]
    const float* bhh  = (const float*)d_in[4];   // [L,T,D]
    const float* watt = (const float*)d_in[5];   // [L,2D]
    const int*   ids  = (const int*)d_in[6];     // [B]
    const int*   nid  = (const int*)d_in[7];     // [T,B,S]
    float* out  = (float*)d_out;                 // [B,D]

    float* agg  = (float*)d_ws;                          // L*T*B*D floats
    float* cur1 = agg + (size_t)Lq * Tq * Bq * Dq;       // B*D floats

    // Both layers' RNN aggregations are independent of `cur` -> one big launch.
    rnn_agg_kernel<<<dim3(Bq / 16, Tq, Lq), 256, 0, stream>>>(
        emb, Wih, Whh, bih, bhh, nid, agg);

    // Layer 0 attention (cur0 gathered from emb[0][ids]) -> cur1 (workspace)
    att_kernel<<<dim3(Bq / 8), 256, 0, stream>>>(
        nullptr, emb, ids, agg, watt, cur1);

    // Layer 1 attention -> final output
    att_kernel<<<dim3(Bq / 8), 256, 0, stream>>>(
        cur1, nullptr, nullptr, agg + (size_t)Tq * Bq * Dq, watt + 2 * Dq, out);
}